// TTime_spatial_25134148616284
// MI455X (gfx1250) — compile-verified
//
#include <hip/hip_runtime.h>
#include <math.h>

typedef __attribute__((ext_vector_type(2))) float v2f;
typedef __attribute__((ext_vector_type(8))) float v8f;

#define NB 512
#define NS 128
#define NK 64
#define XLD 156

__device__ __forceinline__ v8f wmma4(v2f a, v2f b, v8f c) {
  return __builtin_amdgcn_wmma_f32_16x16x4_f32(false, a, false, b, (short)0, c,
                                               false, false);
}

// 32x32 output macro-tile inner loop: two 16-row A fragments from LDS, two
// N-tiles of pre-packed B fragments from global (one b64 per fragment), 2x2
// accumulator quad -> 4 independent WMMA chains per k-step.
__device__ __forceinline__ void gemm_quad(const float* __restrict__ sA, int lda,
                                          int K4, const float* __restrict__ P0,
                                          const float* __restrict__ P1, int ml,
                                          int hh, v8f& c00, v8f& c01, v8f& c10,
                                          v8f& c11) {
  for (int ks = 0; ks < K4; ++ks) {
    int k = ks * 4 + 2 * hh;
    v2f a0, a1, b0, b1;
    a0.x = sA[ml * lda + k];
    a0.y = sA[ml * lda + k + 1];
    a1.x = sA[(16 + ml) * lda + k];
    a1.y = sA[(16 + ml) * lda + k + 1];
    b0 = *(const v2f*)(P0 + ks * 2);
    b1 = *(const v2f*)(P1 + ks * 2);
    c00 = wmma4(a0, b0, c00);
    c01 = wmma4(a0, b1, c01);
    c10 = wmma4(a1, b0, c10);
    c11 = wmma4(a1, b1, c11);
  }
}

__device__ __forceinline__ float wred_sum(float v) {
  for (int o = 16; o > 0; o >>= 1) v += __shfl_xor(v, o, 32);
  return v;
}
__device__ __forceinline__ float wred_max(float v) {
  for (int o = 16; o > 0; o >>= 1) v = fmaxf(v, __shfl_xor(v, o, 32));
  return v;
}

// ---------------- weight repack into WMMA fragment order -----------------
// PW[((nt*32 + lane)*K4 + ks)*2 + j] = W[(4*ks + 2*(lane>>4) + j)*N + nt*16 + (lane&15)]
__global__ void k_pack(const float* __restrict__ W, float* __restrict__ PW,
                       int K, int N) {
  int id = blockIdx.x * blockDim.x + threadIdx.x;
  if (id >= K * N) return;
  int K4 = K >> 2;
  int j = id & 1;
  int t = id >> 1;
  int ks = t % K4;
  t /= K4;
  int lane = t & 31;
  int nt = t >> 5;
  int hh = lane >> 4, ml = lane & 15;
  PW[id] = W[(size_t)(4 * ks + 2 * hh + j) * N + nt * 16 + ml];
}

// ---------------- road lengths / per-b sums ----------------
__global__ __launch_bounds__(128) void k_roadlens(
    const int* __restrict__ roads, const float* __restrict__ ratios,
    const float* __restrict__ lengths, float* __restrict__ RL,
    float* __restrict__ L) {
  int b = blockIdx.x, s = threadIdx.x;
  int lane = s & 31, wave = s >> 5;
  int r = roads[b * NS + s];
  float rl = lengths[r] * ratios[b * NS + s];
  RL[b * NS + s] = rl;
  float sum = wred_sum(rl);
  __shared__ float red[4];
  if (lane == 0) red[wave] = sum;
  __syncthreads();
  if (s == 0) L[b] = red[0] + red[1] + red[2] + red[3];
}

// ---------------- gather + concat embeddings -> X (65536 x 156) ----------
__global__ void k_buildx(const int* __restrict__ roads,
                         const int* __restrict__ mu, const int* __restrict__ m1,
                         const int* __restrict__ m2, const int* __restrict__ m3,
                         const float* __restrict__ eu,
                         const float* __restrict__ e1,
                         const float* __restrict__ e2,
                         const float* __restrict__ e3, float* __restrict__ X) {
  long id = (long)blockIdx.x * blockDim.x + threadIdx.x;
  if (id >= (long)NB * NS * XLD) return;
  int g = (int)(id / XLD);
  int col = (int)(id % XLD);
  int r = roads[g];
  float v;
  if (col < 128)      v = eu[(size_t)mu[r] * 128 + col];
  else if (col < 144) v = e1[m1[r] * 16 + (col - 128)];
  else if (col < 152) v = e2[m2[r] * 8 + (col - 144)];
  else                v = e3[m3[r] * 4 + (col - 152)];
  X[id] = v;
}

// ---------------- conv stack (tiny; direct, BN folded + leaky) -----------
__global__ void k_conv1(const float* __restrict__ T, const float* __restrict__ W,
                        const float* __restrict__ Bc, const float* __restrict__ g,
                        const float* __restrict__ bb, float* __restrict__ O) {
  int id = blockIdx.x * blockDim.x + threadIdx.x;
  if (id >= 32 * 68 * 68) return;
  int ox = id % 68, oy = (id / 68) % 68, oc = id / (68 * 68);
  float acc = 0.f;
  for (int ky = 0; ky < 5; ++ky) {
    int iy = oy * 2 - 1 + ky;
    if ((unsigned)iy >= 138u) continue;
    for (int kx = 0; kx < 5; ++kx) {
      int ix = ox * 2 - 1 + kx;
      if ((unsigned)ix >= 138u) continue;
      acc += T[iy * 138 + ix] * W[oc * 25 + ky * 5 + kx];
    }
  }
  acc += Bc[oc];
  float inv = rsqrtf(1.0f + 1e-5f);
  float y = acc * (g[oc] * inv) + bb[oc];
  O[id] = y > 0.f ? y : 0.1f * y;
}

__global__ void k_conv2(const float* __restrict__ I, const float* __restrict__ W,
                        const float* __restrict__ Bc, const float* __restrict__ g,
                        const float* __restrict__ bb, float* __restrict__ O) {
  int id = blockIdx.x * blockDim.x + threadIdx.x;
  if (id >= 64 * 34 * 34) return;
  int ox = id % 34, oy = (id / 34) % 34, oc = id / (34 * 34);
  float acc = 0.f;
  for (int ic = 0; ic < 32; ++ic)
    for (int ky = 0; ky < 4; ++ky) {
      int iy = oy * 2 - 1 + ky;
      if ((unsigned)iy >= 68u) continue;
      for (int kx = 0; kx < 4; ++kx) {
        int ix = ox * 2 - 1 + kx;
        if ((unsigned)ix >= 68u) continue;
        acc += I[ic * 68 * 68 + iy * 68 + ix] *
               W[oc * 512 + ic * 16 + ky * 4 + kx];
      }
    }
  acc += Bc[oc];
  float inv = rsqrtf(1.0f + 1e-5f);
  float y = acc * (g[oc] * inv) + bb[oc];
  O[id] = y > 0.f ? y : 0.1f * y;
}

__global__ void k_conv3(const float* __restrict__ I, const float* __restrict__ W,
                        const float* __restrict__ Bc, const float* __restrict__ g,
                        const float* __restrict__ bb, float* __restrict__ O) {
  int id = blockIdx.x * blockDim.x + threadIdx.x;
  if (id >= 128 * 17 * 17) return;
  int ox = id % 17, oy = (id / 17) % 17, oc = id / (17 * 17);
  float acc = 0.f;
  for (int ic = 0; ic < 64; ++ic)
    for (int ky = 0; ky < 4; ++ky) {
      int iy = oy * 2 - 1 + ky;
      if ((unsigned)iy >= 34u) continue;
      for (int kx = 0; kx < 4; ++kx) {
        int ix = ox * 2 - 1 + kx;
        if ((unsigned)ix >= 34u) continue;
        acc += I[ic * 34 * 34 + iy * 34 + ix] *
               W[oc * 1024 + ic * 16 + ky * 4 + kx];
      }
    }
  acc += Bc[oc];
  float inv = rsqrtf(1.0f + 1e-5f);
  float y = acc * (g[oc] * inv) + bb[oc];
  O[id] = y > 0.f ? y : 0.1f * y;
}

// ---------------- grid gather + mean over K cells ------------------------
__global__ __launch_bounds__(128) void k_cmean(const float* __restrict__ T3,
                                               const int* __restrict__ lat,
                                               const int* __restrict__ lon,
                                               float* __restrict__ CM) {
  int b = blockIdx.x, c = threadIdx.x;  // c = channel 0..127
  float acc = 0.f;
  for (int k = 0; k < NK; ++k) {
    int p = lat[b * NK + k] * 17 + lon[b * NK + k];
    acc += T3[c * 289 + p];
  }
  CM[b * 128 + c] = acc * (1.0f / 64.0f);
}

// ---------------- f2 MLP: c2 = selu(cm@W1+b1)@W2+b2 (WMMA, M=32) --------
__global__ __launch_bounds__(128) void k_f2(
    const float* __restrict__ CM, const float* __restrict__ PW1,
    const float* __restrict__ B1, const float* __restrict__ PW2,
    const float* __restrict__ B2, float* __restrict__ C2) {
  __shared__ float sC[32 * 128];  // 16KB
  __shared__ float sH[32 * 256];  // 32KB
  int tid = threadIdx.x, lane = tid & 31, wave = tid >> 5;
  int ml = lane & 15, hh = lane >> 4;
  int row0 = blockIdx.x * 32;
  for (int i = tid; i < 32 * 128; i += 128) sC[i] = CM[(size_t)row0 * 128 + i];
  __syncthreads();
  const float SA = 1.6732632423543772f, SS = 1.0507009873554805f;
  // GEMM1: 32x256, K=128 (K4=32); wave covers N-tiles 4w..4w+3 in pairs
  for (int t = 0; t < 4; t += 2) {
    int nt0 = wave * 4 + t, nt1 = nt0 + 1;
    v8f c00 = {}, c01 = {}, c10 = {}, c11 = {};
    gemm_quad(sC, 128, 32, PW1 + (size_t)(nt0 * 32 + lane) * 32 * 2,
              PW1 + (size_t)(nt1 * 32 + lane) * 32 * 2, ml, hh, c00, c01, c10,
              c11);
    int n0 = nt0 * 16 + ml, n1 = nt1 * 16 + ml;
    float bi0 = B1[n0], bi1 = B1[n1];
    for (int v = 0; v < 8; ++v) {
      int r = v + 8 * hh;
      float x;
      x = c00[v] + bi0; sH[r * 256 + n0]        = x > 0.f ? SS * x : SS * SA * (expf(x) - 1.f);
      x = c01[v] + bi1; sH[r * 256 + n1]        = x > 0.f ? SS * x : SS * SA * (expf(x) - 1.f);
      x = c10[v] + bi0; sH[(16 + r) * 256 + n0] = x > 0.f ? SS * x : SS * SA * (expf(x) - 1.f);
      x = c11[v] + bi1; sH[(16 + r) * 256 + n1] = x > 0.f ? SS * x : SS * SA * (expf(x) - 1.f);
    }
  }
  __syncthreads();
  // GEMM2: 32x128, K=256 (K4=64); wave covers N-tiles 2w, 2w+1
  {
    int nt0 = wave * 2, nt1 = nt0 + 1;
    v8f c00 = {}, c01 = {}, c10 = {}, c11 = {};
    gemm_quad(sH, 256, 64, PW2 + (size_t)(nt0 * 32 + lane) * 64 * 2,
              PW2 + (size_t)(nt1 * 32 + lane) * 64 * 2, ml, hh, c00, c01, c10,
              c11);
    int n0 = nt0 * 16 + ml, n1 = nt1 * 16 + ml;
    for (int v = 0; v < 8; ++v) {
      int r = v + 8 * hh;
      C2[(size_t)(row0 + r) * 128 + n0]      = c00[v] + B2[n0];
      C2[(size_t)(row0 + r) * 128 + n1]      = c01[v] + B2[n1];
      C2[(size_t)(row0 + 16 + r) * 128 + n0] = c10[v] + B2[n0];
      C2[(size_t)(row0 + 16 + r) * 128 + n1] = c11[v] + B2[n1];
    }
  }
}

// ---------------- cb = c2 @ f_w1[128:,:] + f_b1 (WMMA, M=32) -------------
__global__ __launch_bounds__(128) void k_cb(const float* __restrict__ C2,
                                            const float* __restrict__ PW1B,
                                            const float* __restrict__ FB1,
                                            float* __restrict__ CB) {
  __shared__ float sC[32 * 128];
  int tid = threadIdx.x, lane = tid & 31, wave = tid >> 5;
  int ml = lane & 15, hh = lane >> 4;
  int row0 = blockIdx.x * 32;
  for (int i = tid; i < 32 * 128; i += 128) sC[i] = C2[(size_t)row0 * 128 + i];
  __syncthreads();
  for (int t = 0; t < 8; t += 2) {  // 32 N-tiles, wave covers 8 in pairs
    int nt0 = wave * 8 + t, nt1 = nt0 + 1;
    v8f c00 = {}, c01 = {}, c10 = {}, c11 = {};
    gemm_quad(sC, 128, 32, PW1B + (size_t)(nt0 * 32 + lane) * 32 * 2,
              PW1B + (size_t)(nt1 * 32 + lane) * 32 * 2, ml, hh, c00, c01, c10,
              c11);
    int n0 = nt0 * 16 + ml, n1 = nt1 * 16 + ml;
    for (int v = 0; v < 8; ++v) {
      int r = v + 8 * hh;
      CB[(size_t)(row0 + r) * 512 + n0]      = c00[v] + FB1[n0];
      CB[(size_t)(row0 + r) * 512 + n1]      = c01[v] + FB1[n1];
      CB[(size_t)(row0 + 16 + r) * 512 + n0] = c10[v] + FB1[n0];
      CB[(size_t)(row0 + 16 + r) * 512 + n1] = c11[v] + FB1[n1];
    }
  }
}

// ---------------- fused ProbRho: rho = relu(X@W1+b1)@W2+b2 (M=32) --------
__global__ __launch_bounds__(128) void k_rho(
    const float* __restrict__ X, const float* __restrict__ PW1,
    const float* __restrict__ B1, const float* __restrict__ PW2,
    const float* __restrict__ B2, float* __restrict__ RHO) {
  __shared__ float sA[32 * XLD];  // 19.5KB
  __shared__ float sH[32 * 512];  // 64KB
  int tid = threadIdx.x, lane = tid & 31, wave = tid >> 5;
  int ml = lane & 15, hh = lane >> 4;
  int row0 = blockIdx.x * 32;
  for (int i = tid; i < 32 * XLD; i += 128) sA[i] = X[(size_t)row0 * XLD + i];
  __syncthreads();
  // GEMM1: 32x512, K=156 (K4=39); wave covers N-tiles 8w..8w+7 in pairs
  for (int t = 0; t < 8; t += 2) {
    int nt0 = wave * 8 + t, nt1 = nt0 + 1;
    v8f c00 = {}, c01 = {}, c10 = {}, c11 = {};
    gemm_quad(sA, XLD, 39, PW1 + (size_t)(nt0 * 32 + lane) * 39 * 2,
              PW1 + (size_t)(nt1 * 32 + lane) * 39 * 2, ml, hh, c00, c01, c10,
              c11);
    int n0 = nt0 * 16 + ml, n1 = nt1 * 16 + ml;
    float bi0 = B1[n0], bi1 = B1[n1];
    for (int v = 0; v < 8; ++v) {
      int r = v + 8 * hh;
      float x;
      x = c00[v] + bi0; sH[r * 512 + n0]        = x > 0.f ? x : 0.f;
      x = c01[v] + bi1; sH[r * 512 + n1]        = x > 0.f ? x : 0.f;
      x = c10[v] + bi0; sH[(16 + r) * 512 + n0] = x > 0.f ? x : 0.f;
      x = c11[v] + bi1; sH[(16 + r) * 512 + n1] = x > 0.f ? x : 0.f;
    }
  }
  __syncthreads();
  // GEMM2: 32x128, K=512 (K4=128); wave covers N-tiles 2w, 2w+1
  {
    int nt0 = wave * 2, nt1 = nt0 + 1;
    v8f c00 = {}, c01 = {}, c10 = {}, c11 = {};
    gemm_quad(sH, 512, 128, PW2 + (size_t)(nt0 * 32 + lane) * 128 * 2,
              PW2 + (size_t)(nt1 * 32 + lane) * 128 * 2, ml, hh, c00, c01, c10,
              c11);
    int n0 = nt0 * 16 + ml, n1 = nt1 * 16 + ml;
    for (int v = 0; v < 8; ++v) {
      int r = v + 8 * hh;
      RHO[(size_t)(row0 + r) * 128 + n0]      = c00[v] + B2[n0];
      RHO[(size_t)(row0 + r) * 128 + n1]      = c01[v] + B2[n1];
      RHO[(size_t)(row0 + 16 + r) * 128 + n0] = c10[v] + B2[n0];
      RHO[(size_t)(row0 + 16 + r) * 128 + n1] = c11[v] + B2[n1];
    }
  }
}

// ---------------- fused TravelTime: logm/logv (M=32) ---------------------
__global__ __launch_bounds__(128) void k_ttime(
    const float* __restrict__ RHO, const float* __restrict__ PW1A,
    const float* __restrict__ CB, const float* __restrict__ w21,
    const float* __restrict__ b21, const float* __restrict__ w22,
    const float* __restrict__ b22, float* __restrict__ LOGM,
    float* __restrict__ LOGV) {
  __shared__ float sR[32 * 128];  // 16KB
  __shared__ float sH[32 * 512];  // 64KB
  int tid = threadIdx.x, lane = tid & 31, wave = tid >> 5;
  int ml = lane & 15, hh = lane >> 4;
  int row0 = blockIdx.x * 32;
  int bidx = row0 >> 7;  // same b for all 32 rows (32 divides S=128)
  for (int i = tid; i < 32 * 128; i += 128) sR[i] = RHO[(size_t)row0 * 128 + i];
  __syncthreads();
  // h3 = relu(rho @ f_w1[:128] + cb[b]); K=128 (K4=32)
  for (int t = 0; t < 8; t += 2) {
    int nt0 = wave * 8 + t, nt1 = nt0 + 1;
    v8f c00 = {}, c01 = {}, c10 = {}, c11 = {};
    gemm_quad(sR, 128, 32, PW1A + (size_t)(nt0 * 32 + lane) * 32 * 2,
              PW1A + (size_t)(nt1 * 32 + lane) * 32 * 2, ml, hh, c00, c01, c10,
              c11);
    int n0 = nt0 * 16 + ml, n1 = nt1 * 16 + ml;
    float cb0 = CB[(size_t)bidx * 512 + n0];
    float cb1 = CB[(size_t)bidx * 512 + n1];
    for (int v = 0; v < 8; ++v) {
      int r = v + 8 * hh;
      float x;
      x = c00[v] + cb0; sH[r * 512 + n0]        = x > 0.f ? x : 0.f;
      x = c01[v] + cb1; sH[r * 512 + n1]        = x > 0.f ? x : 0.f;
      x = c10[v] + cb0; sH[(16 + r) * 512 + n0] = x > 0.f ? x : 0.f;
      x = c11[v] + cb1; sH[(16 + r) * 512 + n1] = x > 0.f ? x : 0.f;
    }
  }
  __syncthreads();
  // logm/logv = h3 . w21/w22 (+b); wave handles 8 rows via shuffle reduce
  for (int rr = 0; rr < 8; ++rr) {
    int r = wave + rr * 4;
    float s1 = 0.f, s2 = 0.f;
    for (int j = lane; j < 512; j += 32) {
      float h = sH[r * 512 + j];
      s1 += h * w21[j];
      s2 += h * w22[j];
    }
    s1 = wred_sum(s1);
    s2 = wred_sum(s2);
    if (lane == 0) {
      LOGM[row0 + r] = s1 + b21[0];
      LOGV[row0 + r] = s2 + b22[0];
    }
  }
}

// ---------------- final log-sum-exp aggregation --------------------------
__global__ __launch_bounds__(128) void k_final(
    const float* __restrict__ RL, const float* __restrict__ L,
    const float* __restrict__ LOGM, const float* __restrict__ LOGV,
    float* __restrict__ out) {
  int b = blockIdx.x, s = threadIdx.x;
  int lane = s & 31, wave = s >> 5;
  __shared__ float redA[4], redB[4];
  float l = L[b];
  float lw = logf(RL[b * NS + s]) - logf(l);
  float a1 = LOGM[b * NS + s] + lw;
  float a2 = LOGV[b * NS + s] + 2.0f * lw;
  float m1 = wred_max(a1), m2 = wred_max(a2);
  if (lane == 0) { redA[wave] = m1; redB[wave] = m2; }
  __syncthreads();
  m1 = fmaxf(fmaxf(redA[0], redA[1]), fmaxf(redA[2], redA[3]));
  m2 = fmaxf(fmaxf(redB[0], redB[1]), fmaxf(redB[2], redB[3]));
  __syncthreads();
  float e1 = wred_sum(expf(a1 - m1));
  float e2 = wred_sum(expf(a2 - m2));
  if (lane == 0) { redA[wave] = e1; redB[wave] = e2; }
  __syncthreads();
  if (s == 0) {
    float lma = m1 + logf(redA[0] + redA[1] + redA[2] + redA[3]);
    float lva = m2 + logf(redB[0] + redB[1] + redB[2] + redB[3]);
    float logl = logf(l);
    out[b] = logl - lma;                   // logmu
    out[NB + b] = logl - 3.f * lma - lva;  // loglam
  }
}

extern "C" void kernel_launch(void* const* d_in, const int* in_sizes, int n_in,
                              void* d_out, int out_size, void* d_ws,
                              size_t ws_size, hipStream_t stream) {
  (void)in_sizes; (void)n_in; (void)out_size; (void)ws_size;
  const int*   roads  = (const int*)d_in[0];
  const float* ratios = (const float*)d_in[1];
  const float* T      = (const float*)d_in[2];
  const int*   lon    = (const int*)d_in[3];
  const int*   lat    = (const int*)d_in[4];
  const float* lens   = (const float*)d_in[5];
  const int*   mu     = (const int*)d_in[6];
  const int*   m1     = (const int*)d_in[7];
  const int*   m2     = (const int*)d_in[8];
  const int*   m3     = (const int*)d_in[9];
  const float* eu     = (const float*)d_in[10];
  const float* e1     = (const float*)d_in[11];
  const float* e2     = (const float*)d_in[12];
  const float* e3     = (const float*)d_in[13];
  const float* rw1    = (const float*)d_in[14];
  const float* rb1    = (const float*)d_in[15];
  const float* rw21   = (const float*)d_in[16];
  const float* rb21   = (const float*)d_in[17];
  const float* c1w    = (const float*)d_in[18];
  const float* c1b    = (const float*)d_in[19];
  const float* g1     = (const float*)d_in[20];
  const float* bb1    = (const float*)d_in[21];
  const float* c2w    = (const float*)d_in[22];
  const float* c2b    = (const float*)d_in[23];
  const float* g2     = (const float*)d_in[24];
  const float* bb2    = (const float*)d_in[25];
  const float* c3w    = (const float*)d_in[26];
  const float* c3b    = (const float*)d_in[27];
  const float* g3     = (const float*)d_in[28];
  const float* bb3    = (const float*)d_in[29];
  const float* f2w1   = (const float*)d_in[30];
  const float* f2b1   = (const float*)d_in[31];
  const float* f2w2   = (const float*)d_in[32];
  const float* f2b2   = (const float*)d_in[33];
  const float* fw1    = (const float*)d_in[34];
  const float* fb1    = (const float*)d_in[35];
  const float* fw21   = (const float*)d_in[36];
  const float* fb21   = (const float*)d_in[37];
  const float* fw22   = (const float*)d_in[38];
  const float* fb22   = (const float*)d_in[39];

  // workspace layout (floats)
  float* ws    = (float*)d_ws;
  float* RL    = ws;                          // 65536
  float* L     = RL + 65536;                  // 512
  float* LOGM  = L + 512;                     // 65536
  float* LOGV  = LOGM + 65536;                // 65536
  float* X     = LOGV + 65536;                // 65536*156
  float* RHO   = X + (size_t)65536 * XLD;     // 65536*128
  float* T1    = RHO + (size_t)65536 * 128;   // 32*68*68
  float* T2    = T1 + 32 * 68 * 68;           // 64*34*34
  float* T3    = T2 + 64 * 34 * 34;           // 128*17*17
  float* CM    = T3 + 128 * 17 * 17;          // 512*128
  float* C2    = CM + 512 * 128;              // 512*128
  float* CB    = C2 + 512 * 128;              // 512*512
  float* PRW1  = CB + 512 * 512;              // 156*512
  float* PRW21 = PRW1 + 156 * 512;            // 512*128
  float* PFW1A = PRW21 + 512 * 128;           // 128*512
  float* PFW1B = PFW1A + 128 * 512;           // 128*512
  float* PF2W1 = PFW1B + 128 * 512;           // 128*256
  float* PF2W2 = PF2W1 + 128 * 256;           // 256*128

  // weight repacks (tiny, L2-resident afterwards)
  hipLaunchKernelGGL(k_pack, dim3((156 * 512 + 255) / 256), dim3(256), 0,
                     stream, rw1, PRW1, 156, 512);
  hipLaunchKernelGGL(k_pack, dim3((512 * 128 + 255) / 256), dim3(256), 0,
                     stream, rw21, PRW21, 512, 128);
  hipLaunchKernelGGL(k_pack, dim3((128 * 512 + 255) / 256), dim3(256), 0,
                     stream, fw1, PFW1A, 128, 512);
  hipLaunchKernelGGL(k_pack, dim3((128 * 512 + 255) / 256), dim3(256), 0,
                     stream, fw1 + (size_t)128 * 512, PFW1B, 128, 512);
  hipLaunchKernelGGL(k_pack, dim3((128 * 256 + 255) / 256), dim3(256), 0,
                     stream, f2w1, PF2W1, 128, 256);
  hipLaunchKernelGGL(k_pack, dim3((256 * 128 + 255) / 256), dim3(256), 0,
                     stream, f2w2, PF2W2, 256, 128);

  long nx = (long)NB * NS * XLD;
  hipLaunchKernelGGL(k_roadlens, dim3(NB), dim3(128), 0, stream, roads, ratios,
                     lens, RL, L);
  hipLaunchKernelGGL(k_buildx, dim3((unsigned)((nx + 255) / 256)), dim3(256), 0,
                     stream, roads, mu, m1, m2, m3, eu, e1, e2, e3, X);
  hipLaunchKernelGGL(k_conv1, dim3((32 * 68 * 68 + 255) / 256), dim3(256), 0,
                     stream, T, c1w, c1b, g1, bb1, T1);
  hipLaunchKernelGGL(k_conv2, dim3((64 * 34 * 34 + 255) / 256), dim3(256), 0,
                     stream, T1, c2w, c2b, g2, bb2, T2);
  hipLaunchKernelGGL(k_conv3, dim3((128 * 17 * 17 + 255) / 256), dim3(256), 0,
                     stream, T2, c3w, c3b, g3, bb3, T3);
  hipLaunchKernelGGL(k_cmean, dim3(NB), dim3(128), 0, stream, T3, lat, lon, CM);
  hipLaunchKernelGGL(k_f2, dim3(NB / 32), dim3(128), 0, stream, CM, PF2W1,
                     f2b1, PF2W2, f2b2, C2);
  hipLaunchKernelGGL(k_cb, dim3(NB / 32), dim3(128), 0, stream, C2, PFW1B, fb1,
                     CB);
  hipLaunchKernelGGL(k_rho, dim3(NB * NS / 32), dim3(128), 0, stream, X, PRW1,
                     rb1, PRW21, rb21, RHO);
  hipLaunchKernelGGL(k_ttime, dim3(NB * NS / 32), dim3(128), 0, stream, RHO,
                     PFW1A, CB, fw21, fb21, fw22, fb22, LOGM, LOGV);
  hipLaunchKernelGGL(k_final, dim3(NB), dim3(128), 0, stream, RL, L, LOGM,
                     LOGV, (float*)d_out);
}